// Sal_seq_8091718385960
// MI455X (gfx1250) — compile-verified
//
#include <hip/hip_runtime.h>
#include <hip/hip_bf16.h>
#include <math.h>

#define IN_F  2048
#define HID   512
#define SEQ_T 14
#define BATCH 256
#define HWPX  49
#define MROWS (BATCH * SEQ_T)   // 3584
#define NG    (4 * HID)         // 2048

typedef __attribute__((ext_vector_type(2))) float v2f;
typedef __attribute__((ext_vector_type(8))) float v8f;

// D = A(16x4 f32) * B(4x16 f32) + C(16x16 f32), full fp32 — matches reference numerics.
__device__ __forceinline__ v8f wmma4(v2f a, v2f b, v8f c) {
  return __builtin_amdgcn_wmma_f32_16x16x4_f32(false, a, false, b, (short)0, c, false, false);
}

__device__ __forceinline__ float sigmoidf_(float x) { return 1.0f / (1.0f + __expf(-x)); }

// ---------- small prep kernels ----------
__global__ void k_lengths(const int* __restrict__ fix, int* __restrict__ len) {
  int b = blockIdx.x * blockDim.x + threadIdx.x;
  if (b < BATCH) {
    int cnt = 0;
    for (int t = 0; t < SEQ_T; ++t) cnt += (fix[b * SEQ_T + t] == 0);
    len[b] = SEQ_T - cnt;
  }
}

__global__ void k_zero(float* __restrict__ p, int n) {
  int i = blockIdx.x * blockDim.x + threadIdx.x;
  if (i < n) p[i] = 0.0f;
}

// Wcat[n][k], n = g*512+h: k<512 -> Wh[g][h][k]; k>=512 -> (g<2 ? Wc[g][h][k-512] : 0)
__global__ void k_wcat(const float* __restrict__ Wh, const float* __restrict__ Wc,
                       float* __restrict__ Wcat) {
  int n = blockIdx.x;
  int g = n >> 9, h = n & 511;
  for (int k = threadIdx.x; k < 1024; k += blockDim.x) {
    float v;
    if (k < HID) v = Wh[(g * HID + h) * HID + k];
    else         v = (g < 2) ? Wc[(g * HID + h) * HID + (k - HID)] : 0.0f;
    Wcat[n * 1024 + k] = v;
  }
}

// ---------- GEMM 1: xproj[m][n] = sum_f x[b][f][fix[m]] * Wx[n][f] + bx[n], m=b*T+t ----------
__global__ __launch_bounds__(256) void k_gemm_xproj(
    const float* __restrict__ x, const int* __restrict__ fix,
    const float* __restrict__ Wx, const float* __restrict__ bx,
    float* __restrict__ xproj) {
  const int lane = threadIdx.x & 31;
  const int l15  = lane & 15;
  const int hi   = lane >> 4;
  const int wave = blockIdx.x * 8 + (threadIdx.x >> 5);
  const int ntiles = NG / 32;                 // 64
  const int mt = wave / ntiles;
  const int nt = wave % ntiles;
  const int m0 = mt * 32, n0 = nt * 32;

  const int mA0 = m0 + l15, mA1 = m0 + 16 + l15;
  // gather fused into A loads: A[m][k] = x[b][k][fix[m]]  (stride HWPX between k's)
  const float* pa0 = x + (size_t)(mA0 / SEQ_T) * IN_F * HWPX + fix[mA0] + 2 * hi * HWPX;
  const float* pa1 = x + (size_t)(mA1 / SEQ_T) * IN_F * HWPX + fix[mA1] + 2 * hi * HWPX;
  const float* pb0 = Wx + (n0 + l15) * IN_F + 2 * hi;
  const float* pb1 = Wx + (n0 + 16 + l15) * IN_F + 2 * hi;

  v8f acc00 = {}, acc01 = {}, acc10 = {}, acc11 = {};
  for (int k0 = 0; k0 < IN_F; k0 += 4) {
    v2f a0 = { pa0[k0 * HWPX], pa0[k0 * HWPX + HWPX] };
    v2f a1 = { pa1[k0 * HWPX], pa1[k0 * HWPX + HWPX] };
    v2f b0 = *(const v2f*)(pb0 + k0);
    v2f b1 = *(const v2f*)(pb1 + k0);
    acc00 = wmma4(a0, b0, acc00);
    acc01 = wmma4(a0, b1, acc01);
    acc10 = wmma4(a1, b0, acc10);
    acc11 = wmma4(a1, b1, acc11);
  }
#pragma unroll
  for (int v = 0; v < 8; ++v) {
    int r0 = m0 + v + 8 * hi, r1 = r0 + 16;
    int c0 = n0 + l15, c1 = c0 + 16;
    xproj[r0 * NG + c0] = acc00[v] + bx[c0];
    xproj[r0 * NG + c1] = acc01[v] + bx[c1];
    xproj[r1 * NG + c0] = acc10[v] + bx[c0];
    xproj[r1 * NG + c1] = acc11[v] + bx[c1];
  }
}

// ---------- Stage A (per step): [h|c] x Wcat^T -> all 4 gate preacts; fused i,f,g,next_c ----------
__global__ __launch_bounds__(256) void k_stageA(
    const float* __restrict__ hc, const float* __restrict__ Wcat,
    const float* __restrict__ xproj, const float* __restrict__ bh,
    const float* __restrict__ bc, float* __restrict__ c_next,
    float* __restrict__ opre, int t) {
  const int lane = threadIdx.x & 31;
  const int l15  = lane & 15, hi = lane >> 4;
  const int wave = blockIdx.x * 8 + (threadIdx.x >> 5);
  const int mt = wave >> 5;            // HID/16 = 32 n-tiles
  const int ht = wave & 31;
  const int m0 = mt * 16;
  const int n  = ht * 16 + l15;        // hidden index (constant per lane)

  const float* pa  = hc   + (m0 + l15) * 1024 + 2 * hi;
  const float* pb0 = Wcat + (0 * HID + n) * 1024 + 2 * hi;
  const float* pb1 = Wcat + (1 * HID + n) * 1024 + 2 * hi;
  const float* pb2 = Wcat + (2 * HID + n) * 1024 + 2 * hi;
  const float* pb3 = Wcat + (3 * HID + n) * 1024 + 2 * hi;

  v8f ai = {}, af = {}, ao = {}, ag = {};
  for (int k0 = 0; k0 < 1024; k0 += 4) {
    v2f a  = *(const v2f*)(pa  + k0);
    v2f b0 = *(const v2f*)(pb0 + k0);
    v2f b1 = *(const v2f*)(pb1 + k0);
    v2f b2 = *(const v2f*)(pb2 + k0);
    v2f b3 = *(const v2f*)(pb3 + k0);
    ai = wmma4(a, b0, ai);
    af = wmma4(a, b1, af);
    ao = wmma4(a, b2, ao);
    ag = wmma4(a, b3, ag);
  }
  const float bi = bh[n]           + bc[n];
  const float bf = bh[HID + n]     + bc[HID + n];
  const float bo = bh[2 * HID + n] + bc[2 * HID + n];
  const float bg = bh[3 * HID + n];
#pragma unroll
  for (int v = 0; v < 8; ++v) {
    int m = m0 + v + 8 * hi;                       // batch row
    const float* xp = xproj + (m * SEQ_T + t) * NG + n;
    float pi = ai[v] + xp[0]        + bi;
    float pf = af[v] + xp[HID]      + bf;
    float po = ao[v] + xp[2 * HID]  + bo;
    float pg = ag[v] + xp[3 * HID]  + bg;
    float iv = sigmoidf_(pi);
    float fv = sigmoidf_(pf);
    float gv = tanhf(pg);
    float cp = hc[m * 1024 + HID + n];
    float cn = fv * cp + iv * gv;
    c_next[m * HID + n] = cn;
    opre[m * HID + n]   = po;
  }
}

// ---------- Stage B (per step): next_c x Wc2^T -> o; h = o*next_c; update [h|c]; crop ----------
__global__ __launch_bounds__(256) void k_stageB(
    const float* __restrict__ c_next, const float* __restrict__ Wc,
    const float* __restrict__ opre, const int* __restrict__ len,
    float* __restrict__ hc, float* __restrict__ last, int t) {
  const int lane = threadIdx.x & 31;
  const int l15  = lane & 15, hi = lane >> 4;
  const int wave = blockIdx.x * 8 + (threadIdx.x >> 5);
  const int nt32 = HID / 32;                 // 16
  const int mt = wave / nt32;                // 0..7
  const int nt = wave % nt32;                // 0..15
  const int m0 = mt * 32, n0 = nt * 32;
  const float* Wc2 = Wc + 2 * HID * HID;

  const float* pa0 = c_next + (m0 + l15) * HID + 2 * hi;
  const float* pa1 = c_next + (m0 + 16 + l15) * HID + 2 * hi;
  const float* pb0 = Wc2 + (n0 + l15) * HID + 2 * hi;
  const float* pb1 = Wc2 + (n0 + 16 + l15) * HID + 2 * hi;

  v8f a00 = {}, a01 = {}, a10 = {}, a11 = {};
  for (int k0 = 0; k0 < HID; k0 += 4) {
    v2f a0 = *(const v2f*)(pa0 + k0);
    v2f a1 = *(const v2f*)(pa1 + k0);
    v2f b0 = *(const v2f*)(pb0 + k0);
    v2f b1 = *(const v2f*)(pb1 + k0);
    a00 = wmma4(a0, b0, a00);
    a01 = wmma4(a0, b1, a01);
    a10 = wmma4(a1, b0, a10);
    a11 = wmma4(a1, b1, a11);
  }
  auto epi = [&](int m, int nn, float accv) {
    float o  = sigmoidf_(opre[m * HID + nn] + accv);
    float cn = c_next[m * HID + nn];
    float hv = o * cn;
    hc[m * 1024 + nn]       = hv;
    hc[m * 1024 + HID + nn] = cn;
    // lengths-1 == -1 wraps to T-1 in the reference
    if (t == (len[m] + SEQ_T - 1) % SEQ_T) last[m * HID + nn] = hv;
  };
#pragma unroll
  for (int v = 0; v < 8; ++v) {
    int r0 = m0 + v + 8 * hi, r1 = r0 + 16;
    int c0 = n0 + l15, c1 = c0 + 16;
    epi(r0, c0, a00[v]);
    epi(r0, c1, a01[v]);
    epi(r1, c0, a10[v]);
    epi(r1, c1, a11[v]);
  }
}

// ---------- final: out[b] = sigmoid(last[b,:] . Wd + bd) ----------
__global__ void k_final(const float* __restrict__ last, const float* __restrict__ Wd,
                        const float* __restrict__ bd, float* __restrict__ out) {
  int b    = blockIdx.x * 8 + (threadIdx.x >> 5);
  int lane = threadIdx.x & 31;
  float s = 0.0f;
  for (int k = lane; k < HID; k += 32) s += last[b * HID + k] * Wd[k];
#pragma unroll
  for (int off = 16; off > 0; off >>= 1) s += __shfl_down(s, off);
  if (lane == 0) out[b] = sigmoidf_(s + bd[0]);
}

extern "C" void kernel_launch(void* const* d_in, const int* in_sizes, int n_in,
                              void* d_out, int out_size, void* d_ws, size_t ws_size,
                              hipStream_t stream) {
  (void)in_sizes; (void)n_in; (void)out_size; (void)ws_size;
  const float* x   = (const float*)d_in[0];
  const int*   fix = (const int*)d_in[1];
  const float* Wx  = (const float*)d_in[2];
  const float* bx  = (const float*)d_in[3];
  const float* Wh  = (const float*)d_in[4];
  const float* bh  = (const float*)d_in[5];
  const float* Wc  = (const float*)d_in[6];
  const float* bc  = (const float*)d_in[7];
  const float* Wd  = (const float*)d_in[8];
  const float* bd  = (const float*)d_in[9];
  float* out = (float*)d_out;

  float* ws     = (float*)d_ws;
  size_t o      = 0;
  float* xproj  = ws + o; o += (size_t)MROWS * NG;     // 29.4 MB
  float* Wcat   = ws + o; o += (size_t)NG * 1024;      // 8 MB
  float* hc     = ws + o; o += (size_t)BATCH * 1024;   // [h|c]
  float* c_next = ws + o; o += (size_t)BATCH * HID;
  float* opre   = ws + o; o += (size_t)BATCH * HID;
  float* last   = ws + o; o += (size_t)BATCH * HID;
  int*   len    = (int*)(ws + o);

  k_lengths<<<1, 256, 0, stream>>>(fix, len);
  k_zero<<<(BATCH * 1024 + 255) / 256, 256, 0, stream>>>(hc, BATCH * 1024);
  k_wcat<<<NG, 256, 0, stream>>>(Wh, Wc, Wcat);
  k_gemm_xproj<<<(MROWS / 32) * (NG / 32) / 8, 256, 0, stream>>>(x, fix, Wx, bx, xproj);
  for (int t = 0; t < SEQ_T; ++t) {
    k_stageA<<<(BATCH / 16) * (HID / 16) / 8, 256, 0, stream>>>(hc, Wcat, xproj, bh, bc,
                                                                c_next, opre, t);
    k_stageB<<<(BATCH / 32) * (HID / 32) / 8, 256, 0, stream>>>(c_next, Wc, opre, len,
                                                                hc, last, t);
  }
  k_final<<<BATCH / 8, 256, 0, stream>>>(last, Wd, bd, out);
}